// Block_LightGCNConv_3358664426025
// MI455X (gfx1250) — compile-verified
//
#include <hip/hip_runtime.h>

// ---------------------------------------------------------------------------
// LightGCN-style 2-block weighted gather/scatter-sum for MI455X (gfx1250).
//
// Memory-bound: x (102 MB) and h0 (25.6 MB) are L2-resident (192 MB L2).
// Strategy: one wave32 per edge; async global->LDS DMA (ASYNCcnt) pipelines
// 4 edge-rows per wave; hardware f32 atomics do the segment sum.
// (edge_scatter_kernel is defined FIRST so the disasm snippet shows it.)
// ---------------------------------------------------------------------------

#define DFEAT 256
#define PIPE 4               // edge rows in flight per wave (2 async ops each)
#define WAVES_PER_BLOCK 8
#define BLOCK_THREADS (WAVES_PER_BLOCK * 32)
#define N_DST0 25000

typedef int v4i __attribute__((ext_vector_type(4)));
typedef __attribute__((address_space(1))) v4i* global_v4i_ptr;
typedef __attribute__((address_space(3))) v4i* lds_v4i_ptr;

// ---- CDNA5 async global->LDS copy (ASYNCcnt path) --------------------------
#if __has_builtin(__builtin_amdgcn_s_wait_asynccnt)
#define WAIT_ASYNCCNT(n)                        \
  do {                                          \
    __builtin_amdgcn_s_wait_asynccnt(n);        \
    asm volatile("" ::: "memory");              \
  } while (0)
#else
#define WAIT_ASYNCCNT(n) \
  asm volatile("s_wait_asynccnt %0" ::"i"(n) : "memory")
#endif

// Copy one 1024-byte feature row (256 f32) global -> LDS, per-wave:
// lane L moves bytes [L*16, L*16+16) and [512+L*16, 512+L*16+16).
// gsrc_lane / lds_lane are this lane's base (row + lane*4 floats).
__device__ __forceinline__ void async_copy_row(const float* gsrc_lane,
                                               float* lds_lane) {
#if __has_builtin(__builtin_amdgcn_global_load_async_to_lds_b128)
  global_v4i_ptr g = (global_v4i_ptr)(gsrc_lane);
  lds_v4i_ptr l = (lds_v4i_ptr)(lds_lane);
  __builtin_amdgcn_global_load_async_to_lds_b128(g, l, 0, 0);
  __builtin_amdgcn_global_load_async_to_lds_b128(g, l, 512, 0);
#else
  unsigned lds_off =
      (unsigned)(unsigned long)(__attribute__((address_space(3))) char*)(void*)
          lds_lane;
  unsigned long long gaddr = (unsigned long long)(size_t)gsrc_lane;
  asm volatile(
      "global_load_async_to_lds_b128 %0, %1, off\n\t"
      "global_load_async_to_lds_b128 %0, %1, off offset:512" ::"v"(lds_off),
      "v"(gaddr)
      : "memory");
#endif
}

// One wave per edge: gather x[src[e]] (256 f32) via async DMA into a per-wave
// LDS slot, scale by ew[e], scatter-add into out[dst[e]] with f32 atomics.
__global__ void __launch_bounds__(BLOCK_THREADS) edge_scatter_kernel(
    const float* __restrict__ x, const float* __restrict__ ew,
    const int* __restrict__ src, const int* __restrict__ dst,
    float* __restrict__ out, int E) {
  __shared__ __align__(16) float smem[WAVES_PER_BLOCK][PIPE][DFEAT];

  const int lane = threadIdx.x & 31;
  const int wib = threadIdx.x >> 5;  // wave index in block
  const long long gw = (long long)blockIdx.x * WAVES_PER_BLOCK + wib;
  const long long nw = (long long)gridDim.x * WAVES_PER_BLOCK;

  float* slot_base = &smem[wib][0][0];

  // Prologue: fill the pipeline with up to PIPE edge rows.
  long long e_issue = gw;
  for (int p = 0; p < PIPE; ++p) {
    if (e_issue < E) {
      const int s = __builtin_amdgcn_readfirstlane(src[e_issue]);
      async_copy_row(x + (long long)s * DFEAT + lane * 4,
                     slot_base + p * DFEAT + lane * 4);
      e_issue += nw;
    }
  }

  int it = 0;
  for (long long e = gw; e < E; e += nw, ++it) {
    const int slot = it & (PIPE - 1);

    if (e_issue < (long long)E) {
      // Steady state: PIPE edges (2*PIPE = 8 ops) outstanding; async loads
      // complete in order, so asynccnt <= 6 means the oldest row landed.
      WAIT_ASYNCCNT((PIPE - 1) * 2);
    } else {
      // Drain: fewer than PIPE in flight; wait for everything.
      WAIT_ASYNCCNT(0);
    }

    // Scalar (wave-uniform) edge metadata.
    const int d = __builtin_amdgcn_readfirstlane(dst[e]);
    const float w = __uint_as_float(
        __builtin_amdgcn_readfirstlane(__float_as_uint(ew[e])));

    // Read this wave's slot from LDS: 8 floats per lane.
    const float4 a =
        *(const float4*)(slot_base + slot * DFEAT + lane * 4);
    const float4 b =
        *(const float4*)(slot_base + slot * DFEAT + 128 + lane * 4);

    // Refill the just-consumed slot with the next edge's row.
    if (e_issue < (long long)E) {
      const int s2 = __builtin_amdgcn_readfirstlane(src[e_issue]);
      async_copy_row(x + (long long)s2 * DFEAT + lane * 4,
                     slot_base + slot * DFEAT + lane * 4);
      e_issue += nw;
    }

    float* o = out + (long long)d * DFEAT + lane * 4;
    unsafeAtomicAdd(o + 0, a.x * w);
    unsafeAtomicAdd(o + 1, a.y * w);
    unsafeAtomicAdd(o + 2, a.z * w);
    unsafeAtomicAdd(o + 3, a.w * w);
    unsafeAtomicAdd(o + 128, b.x * w);
    unsafeAtomicAdd(o + 129, b.y * w);
    unsafeAtomicAdd(o + 130, b.z * w);
    unsafeAtomicAdd(o + 131, b.w * w);
  }
}

__global__ void __launch_bounds__(256) zero_f32x4_kernel(float4* p, int n4) {
  int i = blockIdx.x * blockDim.x + threadIdx.x;
  if (i < n4) p[i] = make_float4(0.f, 0.f, 0.f, 0.f);
}

extern "C" void kernel_launch(void* const* d_in, const int* in_sizes, int n_in,
                              void* d_out, int out_size, void* d_ws,
                              size_t ws_size, hipStream_t stream) {
  const float* x = (const float*)d_in[0];
  const float* ew0 = (const float*)d_in[1];
  const float* ew1 = (const float*)d_in[2];
  const int* src0 = (const int*)d_in[3];
  const int* dst0 = (const int*)d_in[4];
  const int* src1 = (const int*)d_in[5];
  const int* dst1 = (const int*)d_in[6];
  float* out = (float*)d_out;
  float* h0 = (float*)d_ws;  // 25000*256 f32 = 25.6 MB scratch

  const int E0 = in_sizes[3];
  const int E1 = in_sizes[5];

  // Zero the intermediate and the output (harness poisons them with 0xAA).
  {
    const int n4 = N_DST0 * DFEAT / 4;
    zero_f32x4_kernel<<<(n4 + 255) / 256, 256, 0, stream>>>((float4*)h0, n4);
  }
  {
    const int n4 = out_size / 4;
    zero_f32x4_kernel<<<(n4 + 255) / 256, 256, 0, stream>>>((float4*)out, n4);
  }

  // Block 0: h0 = segment_sum(x[src0] * ew0, dst0)
  edge_scatter_kernel<<<2048, BLOCK_THREADS, 0, stream>>>(x, ew0, src0, dst0,
                                                          h0, E0);
  // Block 1: out = segment_sum(h0[src1] * ew1, dst1)
  edge_scatter_kernel<<<1024, BLOCK_THREADS, 0, stream>>>(h0, ew1, src1, dst1,
                                                          out, E1);
}